// PostProcessingModule_39943195853061
// MI455X (gfx1250) — compile-verified
//
#include <hip/hip_runtime.h>
#include <cmath>

// ---------------------------------------------------------------------------
// PostProcessingModule for MI455X (gfx1250, wave32).
//
// rows are processed kpos-major: each wave owns 16 consecutive points at one
// window position, loops kpos = 0..48, and carries the 16x32 conv accumulator
// in WMMA D registers across the whole loop (chained V_WMMA_F32_16X16X4_F32).
// fp32 WMMA keeps exact fp32 semantics vs the reference (K=4 == CIN natively).
// ---------------------------------------------------------------------------

typedef float v2f __attribute__((ext_vector_type(2)));
typedef float v8f __attribute__((ext_vector_type(8)));

// Only check on the DEVICE pass: amdgcn builtins are never visible to the
// host-side clang invocation of a HIP TU.
#if defined(__HIP_DEVICE_COMPILE__)
#if !__has_builtin(__builtin_amdgcn_wmma_f32_16x16x4_f32)
#error "gfx1250 device pass: __builtin_amdgcn_wmma_f32_16x16x4_f32 missing"
#endif
#endif

#define HH    64
#define WW    512
#define NN    (HH * WW)      // 32768 points
#define KWIN  7
#define PADK  3
#define CIN   4
#define HID   64
#define PC    32
#define NC    19

#define BLOCK 256            // 8 wave32 waves
#define WAVES 8
#define PTSW  16             // points per wave (one WMMA M-tile)
#define PTSB  (WAVES * PTSW) // 128 points per block

// f32 WMMA: D(16x16) = A(16x4) * B(4x16) + C.  A,B = v2f per lane, C/D = v8f.
__device__ __forceinline__ v8f wmma_f32(v2f a, v2f b, v8f c) {
#if defined(__HIP_DEVICE_COMPILE__)
  return __builtin_amdgcn_wmma_f32_16x16x4_f32(false, a, false, b, (short)0, c,
                                               false, false);
#else
  return c;  // host stub (never executed)
#endif
}

// Butterfly reductions over each 16-lane half of the wave (masks < 16 never
// cross the half boundary). After this every lane of a half holds the result
// for its row (rows M and M+8 live in opposite halves of the WMMA D layout).
__device__ __forceinline__ float hsum16(float v) {
  v += __shfl_xor(v, 1, 32);
  v += __shfl_xor(v, 2, 32);
  v += __shfl_xor(v, 4, 32);
  v += __shfl_xor(v, 8, 32);
  return v;
}
__device__ __forceinline__ float hmax16(float v) {
  v = fmaxf(v, __shfl_xor(v, 1, 32));
  v = fmaxf(v, __shfl_xor(v, 2, 32));
  v = fmaxf(v, __shfl_xor(v, 4, 32));
  v = fmaxf(v, __shfl_xor(v, 8, 32));
  return v;
}

__global__ void zero_f32_kernel(float* __restrict__ p, int n) {
  int i = blockIdx.x * blockDim.x + threadIdx.x;
  if (i < n) p[i] = 0.0f;
}

__global__ __launch_bounds__(BLOCK, 1) void ppm_kernel(
    const float* __restrict__ x,        // (PC, H, W)
    const float* __restrict__ proj,     // (CIN, H, W)
    const float* __restrict__ ub,       // (N, CIN)
    const int*   __restrict__ lut,      // (N, 3) : pt, yy, xx
    const int*   __restrict__ nvp,      // (1,)
    const float* __restrict__ W1,       // (HID, CIN)
    const float* __restrict__ b1,       // (HID,)
    const float* __restrict__ gln,      // (HID,)
    const float* __restrict__ bln,      // (HID,)
    const float* __restrict__ W2,       // (PC, HID)
    const float* __restrict__ b2,       // (PC,)
    const float* __restrict__ Wc,       // (NC, PC, K, K)
    const float* __restrict__ bc,       // (NC,)
    float* __restrict__ out)            // (NC, N)
{
  // ---- LDS (about 63 KB; even float strides so fragment reads are b64) ----
  __shared__ float s_w1[HID][4];            // W_fc1[col][k]
  __shared__ float s_b1[HID], s_g[HID], s_bt[HID];
  __shared__ float s_w2[PC][68];            // W_fc2[col][k], padded
  __shared__ float s_b2[PC];
  __shared__ float s_bc[NC + 1];
  __shared__ float s_wc[PC][20];            // per-kpos W_conv^T slice [c][o]
  __shared__ float s_h[WAVES][16][66];      // per-wave h (16 rows x 64 hid)
  __shared__ float s_p[WAVES][16][34];      // per-wave P (16 rows x 32 ch)

  const int tid = threadIdx.x;

  // stage loop-invariant weights once
  for (int i = tid; i < HID * CIN; i += BLOCK) s_w1[i >> 2][i & 3] = W1[i];
  for (int i = tid; i < HID; i += BLOCK) {
    s_b1[i] = b1[i]; s_g[i] = gln[i]; s_bt[i] = bln[i];
  }
  for (int i = tid; i < PC * HID; i += BLOCK) s_w2[i >> 6][i & 63] = W2[i];
  for (int i = tid; i < PC; i += BLOCK) s_b2[i] = b2[i];
  for (int i = tid; i < NC; i += BLOCK) s_bc[i] = bc[i];
  __syncthreads();

  const int wid  = tid >> 5;
  const int lane = tid & 31;
  const int lm   = lane & 15;      // WMMA: A-row / B,D-column index
  const int hi   = lane >> 4;      // WMMA: half-wave selector
  const int n0   = blockIdx.x * PTSB + wid * PTSW;

  // --- A-side point (rows of the MLP tile): M = lm ---
  const int nA = n0 + lm;
  const int pA = lut[nA * 3 + 0];
  const int yA = lut[nA * 3 + 1];
  const int xA = lut[nA * 3 + 2];
  // feat channels owned by this lane in the A fragment: c = 2*hi + j
  const float cen0 = ub[pA * CIN + 2 * hi + 0];
  const float cen1 = ub[pA * CIN + 2 * hi + 1];

  // --- D-side rows (M = i + 8*hi) for P construction / final scatter ---
  int yD[8], xD[8];
#pragma unroll
  for (int i = 0; i < 8; ++i) {
    int nD = n0 + i + 8 * hi;
    yD[i] = lut[nD * 3 + 1];
    xD[i] = lut[nD * 3 + 2];
  }

  // hoisted fc1 B fragments and per-column constants (col = 16t + lm)
  v2f   bw1[4];
  float b1v[4], gv[4], btv[4];
#pragma unroll
  for (int t = 0; t < 4; ++t) {
    int col = 16 * t + lm;
    bw1[t] = *(const v2f*)&s_w1[col][2 * hi];  // B[k][col], k = 2*hi + j
    b1v[t] = s_b1[col]; gv[t] = s_g[col]; btv[t] = s_bt[col];
  }
  const float b2v0 = s_b2[lm], b2v1 = s_b2[16 + lm];

  v8f cacc0 = {}; v8f cacc1 = {};   // conv accumulators, chained over kpos
  const float INV_SQRT2 = 0.70710678118654752f;

  for (int kpos = 0; kpos < KWIN * KWIN; ++kpos) {
    const int ki = kpos / KWIN, kj = kpos % KWIN;

    // stage W_conv^T slice for this kpos: s_wc[c][o] = Wc[o][c][ki][kj]
    __syncthreads();  // previous iteration's conv reads are done
    for (int i = tid; i < PC * NC; i += BLOCK) {
      int c = i / NC, o = i % NC;
      s_wc[c][o] = Wc[(o * PC + c) * (KWIN * KWIN) + kpos];
    }
    __syncthreads();

    // -------- feat: A fragment (16x4), lane element (M=lm, K=2*hi+j) ------
    v2f afrag;
    {
      int y = yA + ki - PADK, xc2 = xA + kj - PADK;
      bool vld = (y >= 0) & (y < HH) & (xc2 >= 0) & (xc2 < WW);
      int yc = min(max(y, 0), HH - 1), xc = min(max(xc2, 0), WW - 1);
      int base = yc * WW + xc;
      float v0 = proj[(2 * hi + 0) * NN + base];
      float v1 = proj[(2 * hi + 1) * NN + base];
      v0 = vld ? v0 : 0.0f;   // zero-padded window
      v1 = vld ? v1 : 0.0f;
      afrag[0] = fabsf(v0 - cen0);
      afrag[1] = fabsf(v1 - cen1);
      // pull next kpos's proj row toward the caches
      int kn = min(kpos + 1, KWIN * KWIN - 1);
      int yn = min(max(yA + kn / KWIN - PADK, 0), HH - 1);
      int xn = min(max(xA + kn % KWIN - PADK, 0), WW - 1);
      __builtin_prefetch(&proj[2 * hi * NN + yn * WW + xn], 0, 1);
    }

    // -------- fc1: h(16x64) = feat @ W1^T, four 16-col tiles --------------
    v8f hT[4];
#pragma unroll
    for (int t = 0; t < 4; ++t) {
      v8f z = {};
      hT[t] = wmma_f32(afrag, bw1[t], z);
    }

    // -------- +bias, LayerNorm(64) + exact GELU, store rows to LDS --------
#pragma unroll
    for (int t = 0; t < 4; ++t)
#pragma unroll
      for (int i = 0; i < 8; ++i) hT[t][i] += b1v[t];

#pragma unroll
    for (int i = 0; i < 8; ++i) {
      float s = hT[0][i] + hT[1][i] + hT[2][i] + hT[3][i];
      float q = hT[0][i] * hT[0][i] + hT[1][i] * hT[1][i] +
                hT[2][i] * hT[2][i] + hT[3][i] * hT[3][i];
      s = hsum16(s);                       // row sum over 64 hidden units
      q = hsum16(q);
      float mu   = s * (1.0f / HID);
      float var  = q * (1.0f / HID) - mu * mu;
      float rstd = rsqrtf(var + 1e-5f);
      int M = i + 8 * hi;
#pragma unroll
      for (int t = 0; t < 4; ++t) {
        float v = (hT[t][i] - mu) * rstd * gv[t] + btv[t];
        v = 0.5f * v * (1.0f + erff(v * INV_SQRT2));   // exact GELU
        s_h[wid][M][16 * t + lm] = v;
      }
    }
    __builtin_amdgcn_wave_barrier();   // wave-private LDS phase boundary

    // -------- fc2: (16x64)@(64x32), 16 chained K=4 WMMA steps -------------
    v8f o0 = {}, o1 = {};
#pragma unroll
    for (int s = 0; s < 16; ++s) {
      v2f a2 = *(const v2f*)&s_h[wid][lm][4 * s + 2 * hi];       // A[M=lm][k]
      v2f bA = *(const v2f*)&s_w2[lm][4 * s + 2 * hi];           // B[k][col]
      v2f bB = *(const v2f*)&s_w2[16 + lm][4 * s + 2 * hi];
      o0 = wmma_f32(a2, bA, o0);
      o1 = wmma_f32(a2, bB, o1);
    }

    // -------- softmax(32) and P = wgt * win_x, store rows to LDS ----------
#pragma unroll
    for (int i = 0; i < 8; ++i) { o0[i] += b2v0; o1[i] += b2v1; }
#pragma unroll
    for (int i = 0; i < 8; ++i) {
      float m  = hmax16(fmaxf(o0[i], o1[i]));
      float e0 = expf(o0[i] - m), e1 = expf(o1[i] - m);
      float inv = 1.0f / hsum16(e0 + e1);
      int M = i + 8 * hi;
      int y = yD[i] + ki - PADK, xc2 = xD[i] + kj - PADK;
      bool vld = (y >= 0) & (y < HH) & (xc2 >= 0) & (xc2 < WW);
      int yc = min(max(y, 0), HH - 1), xc = min(max(xc2, 0), WW - 1);
      int base = yc * WW + xc;
      float w0 = x[lm * NN + base];          // channel c = lm      (tile 0)
      float w1 = x[(16 + lm) * NN + base];   // channel c = 16 + lm (tile 1)
      w0 = vld ? w0 : 0.0f;
      w1 = vld ? w1 : 0.0f;
      s_p[wid][M][lm]      = (e0 * inv) * w0;
      s_p[wid][M][16 + lm] = (e1 * inv) * w1;
    }
    __builtin_amdgcn_wave_barrier();

    // -------- conv: acc += P(16x32) @ W_conv^T(32x19), 8 K-steps ----------
#pragma unroll
    for (int s = 0; s < 8; ++s) {
      v2f a3 = *(const v2f*)&s_p[wid][lm][4 * s + 2 * hi];
      int c0 = 4 * s + 2 * hi, c1 = c0 + 1;
      v2f bC0, bC1;
      bC0[0] = s_wc[c0][lm];                 // o = lm (< 19 always)
      bC0[1] = s_wc[c1][lm];
      int o1i = 16 + lm;
      int ocl = min(o1i, NC - 1);
      float g0 = s_wc[c0][ocl], g1 = s_wc[c1][ocl];
      bC1[0] = (o1i < NC) ? g0 : 0.0f;       // pad classes 19..31 with zeros
      bC1[1] = (o1i < NC) ? g1 : 0.0f;
      cacc0 = wmma_f32(a3, bC0, cacc0);
      cacc1 = wmma_f32(a3, bC1, cacc1);
    }
  } // kpos

  // -------- finalize: + b_conv, validity mask, scatter-add at pt ----------
  const int nv = nvp[0];
#pragma unroll
  for (int i = 0; i < 8; ++i) {
    int M  = i + 8 * hi;
    int nD = n0 + M;
    int pt = lut[nD * 3 + 0];
    bool rowok = (nD < nv);
    if (rowok) {
      atomicAdd(&out[lm * NN + pt], cacc0[i] + s_bc[lm]);
      int o1i = 16 + lm;
      if (o1i < NC) atomicAdd(&out[o1i * NN + pt], cacc1[i] + s_bc[o1i]);
    }
  }
}

extern "C" void kernel_launch(void* const* d_in, const int* in_sizes, int n_in,
                              void* d_out, int out_size, void* d_ws,
                              size_t ws_size, hipStream_t stream) {
  (void)in_sizes; (void)n_in; (void)d_ws; (void)ws_size;
  const float* x    = (const float*)d_in[0];
  const float* proj = (const float*)d_in[1];
  const float* ub   = (const float*)d_in[2];
  const int*   lut  = (const int*)d_in[3];
  const int*   nvp  = (const int*)d_in[4];
  const float* W1   = (const float*)d_in[5];
  const float* b1   = (const float*)d_in[6];
  const float* gln  = (const float*)d_in[7];
  const float* bln  = (const float*)d_in[8];
  const float* W2   = (const float*)d_in[9];
  const float* b2   = (const float*)d_in[10];
  const float* Wc   = (const float*)d_in[11];
  const float* bc   = (const float*)d_in[12];
  float* out = (float*)d_out;

  const int total = NC * NN;
  zero_f32_kernel<<<(total + BLOCK - 1) / BLOCK, BLOCK, 0, stream>>>(out, total);
  ppm_kernel<<<NN / PTSB, BLOCK, 0, stream>>>(x, proj, ub, lut, nvp, W1, b1,
                                              gln, bln, W2, b2, Wc, bc, out);
}